// sparselinear_layer_91147795956228
// MI455X (gfx1250) — compile-verified
//
#include <hip/hip_runtime.h>

// CDNA5 / gfx1250, wave32.
typedef __attribute__((ext_vector_type(2))) float v2f;
typedef __attribute__((ext_vector_type(8))) float v8f;

#define WAVE 32
#define BATCH 64   // B fixed by the reference problem

// ---------------------------------------------------------------------------
// Stage 1: transpose x [64, G] (row-major) -> xT [G, 64] so that a per-edge
// gather of x[:, c] becomes one contiguous 256B read. LDS-tiled: 64 b x 32 g.
// ---------------------------------------------------------------------------
__global__ void xpose_x_kernel(const float* __restrict__ x,
                               float* __restrict__ xT, int G) {
    __shared__ float tile[BATCH][33];   // +1 pad: avoid bank conflicts
    const int g0 = blockIdx.x * 32;
    const int tx = threadIdx.x & 31;    // g within tile
    const int ty = threadIdx.x >> 5;    // 0..7
    const int g  = g0 + tx;
    if (g < G) {
        #pragma unroll
        for (int bb = ty; bb < BATCH; bb += 8)
            tile[bb][tx] = x[(size_t)bb * G + g];
    }
    __syncthreads();
    const int bl = threadIdx.x & 63;    // b (coalesced on write)
    const int gq = threadIdx.x >> 6;    // 0..3
    #pragma unroll
    for (int gl = gq; gl < 32; gl += 4) {
        const int gg = g0 + gl;
        if (gg < G) xT[(size_t)gg * BATCH + bl] = tile[bl][gl];
    }
}

// ---------------------------------------------------------------------------
// Stage 3: one wave per COO edge. Lane l handles batch elements 2l, 2l+1.
//   load  : xT[c*64 + 2l .. 2l+1]   -> one coalesced 256B wave read (L2-hot)
//   atomic: acc[r*64 + 2l .. 2l+1]  -> contiguous 256B of f32 atomic adds
// This is the canonical segment_sum; avg 10 edges/row => low contention.
// ---------------------------------------------------------------------------
__global__ void edge_scatter_kernel(const int* __restrict__ rows,
                                    const int* __restrict__ cols,
                                    const float* __restrict__ w,
                                    const float* __restrict__ xT,
                                    float* __restrict__ acc, int nnz) {
    const int wave = (int)((blockIdx.x * (size_t)blockDim.x + threadIdx.x) >> 5);
    const int lane = threadIdx.x & 31;
    if (wave >= nnz) return;
    const int   r  = rows[wave];      // broadcast load (all lanes same line)
    const int   c  = cols[wave];
    const float wt = w[wave];
    const v2f xv = ((const v2f*)xT)[(size_t)c * (BATCH / 2) + lane];
    float* dst = acc + (size_t)r * BATCH + 2 * lane;
    atomicAdd(dst,     wt * xv.x);
    atomicAdd(dst + 1, wt * xv.y);
}

// ---------------------------------------------------------------------------
// Stage 4: transpose acc [S,64] -> out [64,S] and add bias, using
// V_WMMA_F32_16X16X4_F32 as an in-register transpose engine.
//
// For a 16x16 tile M (rows = r, cols = b):  D = sum_k A_k x B_k,  where
//   A_k(i,kk) = M[4k+kk, i]   (tile^T column chunk; coalesced row loads)
//   B_k       = I[4k:4k+4, :] (identity slice, built in registers)
// => D = M^T.  Seeding the accumulator C with bias[r] folds the bias in free.
//
// f32 16x16x4 operand layouts (ISA 7.12.2):
//   A: lane L<16 holds A(L,0),A(L,1); lane L>=16 holds A(L-16,2),A(L-16,3)
//   B: lane n<16 holds B(0,n),B(1,n); lane n>=16 holds B(2,n-16),B(3,n-16)
//   D: vgpr v, lane n: D[v + 8*(n>=16)][n%16]
// ---------------------------------------------------------------------------
__global__ void wmma_xpose_bias_kernel(const float* __restrict__ acc,
                                       const float* __restrict__ bias,
                                       float* __restrict__ out, int S) {
    const int wave = (int)((blockIdx.x * (size_t)blockDim.x + threadIdx.x) >> 5);
    const int lane = threadIdx.x & 31;
    const int tile = wave;                 // (S/16) * 4 tiles total
    const int r0 = (tile >> 2) * 16;       // 4 b-tiles per r-range (B=64)
    const int b0 = (tile & 3) * 16;
    if (r0 + 16 > S) return;               // uniform per wave: EXEC stays full

    const int half = lane >> 4;            // 0 | 1
    const int sub  = lane & 15;            // 0..15

    // C = bias[r0 + n] broadcast across all 8 accumulator VGPRs (n = r index)
    const float bv = bias[r0 + sub];
    v8f c;
    #pragma unroll
    for (int i = 0; i < 8; ++i) c[i] = bv;

    #pragma unroll
    for (int k = 0; k < 4; ++k) {
        const int rr = 4 * k + 2 * half;   // M-row this half-wave sources
        v2f a, bm;
        a.x = acc[(size_t)(r0 + rr)     * BATCH + b0 + sub];
        a.y = acc[(size_t)(r0 + rr + 1) * BATCH + b0 + sub];
        bm.x = (sub == rr)     ? 1.0f : 0.0f;   // identity slice I[4k:4k+4,:]
        bm.y = (sub == rr + 1) ? 1.0f : 0.0f;
        c = __builtin_amdgcn_wmma_f32_16x16x4_f32(
                /*neg_a=*/false, a, /*neg_b=*/false, bm,
                /*c_mod=*/(short)0, c, /*reuse_a=*/false, /*reuse_b=*/false);
    }

    // D[v + 8*half][sub] = out element (b = b0+v+8*half, r = r0+sub);
    // each half-wave writes a contiguous 64B run along r.
    #pragma unroll
    for (int v = 0; v < 8; ++v) {
        const int b = b0 + v + 8 * half;
        out[(size_t)b * S + r0 + sub] = c[v];
    }
}

// ---------------------------------------------------------------------------
extern "C" void kernel_launch(void* const* d_in, const int* in_sizes, int n_in,
                              void* d_out, int out_size, void* d_ws, size_t ws_size,
                              hipStream_t stream) {
    // setup_inputs order: x, W_sparse, b, index, n_SNPs
    const float* x   = (const float*)d_in[0];
    const float* W   = (const float*)d_in[1];
    const float* b   = (const float*)d_in[2];
    const int*   idx = (const int*)d_in[3];   // [2, NNZ]; harness passes int32

    const int G   = in_sizes[0] / BATCH;      // 20000
    const int nnz = in_sizes[3] / 2;          // 2,000,000
    const int S   = out_size / BATCH;         // 200000 (avoids host-reading n_SNPs)
    float* out = (float*)d_out;

    // Workspace layout: xT [G,64] | acc [S,64]  (~5.1 MB + ~51.2 MB)
    char* ws = (char*)d_ws;
    float* xT = (float*)ws;
    size_t accOff = (((size_t)G * BATCH * sizeof(float)) + 255) & ~(size_t)255;
    float* acc = (float*)(ws + accOff);

    // Stage 2: zero the atomic accumulator (async memset is graph-capturable)
    hipMemsetAsync(acc, 0, (size_t)S * BATCH * sizeof(float), stream);

    // Stage 1: x -> xT
    xpose_x_kernel<<<(G + 31) / 32, 256, 0, stream>>>(x, xT, G);

    // Stage 3: one wave per edge
    {
        long long threads = (long long)nnz * WAVE;
        int blocks = (int)((threads + 255) / 256);
        edge_scatter_kernel<<<blocks, 256, 0, stream>>>(idx, idx + nnz, W, xT,
                                                        acc, nnz);
    }

    // Stage 4: WMMA transpose + bias, 8 tiles (waves) per 256-thread block
    {
        int tiles  = ((S + 15) / 16) * 4;
        int blocks = (tiles + 7) / 8;
        wmma_xpose_bias_kernel<<<blocks, 256, 0, stream>>>(acc, b, out, S);
    }
}